// CatmullRomActivation_1271310319658
// MI455X (gfx1250) — compile-verified
//
#include <hip/hip_runtime.h>

// Catmull-Rom spline activation for MI455X (gfx1250, wave32).
// Memory-bound elementwise op: stream x -> out with 128-bit NT vector
// accesses; stage the hot per-neuron control-point table in LDS (padded
// stride to reduce bank conflicts) and gather the 4 spline points from LDS.

#define M_ROWS     8192
#define N_NEURONS  2048
#define KCP        10          // control points per neuron (incl. ghosts)
#define CHUNK_N    512         // neurons per block  (N_NEURONS / gridDim.x)
#define ROWS_PB    32          // rows per block     (M_ROWS   / gridDim.y)
#define LDS_STRIDE 11          // padded row stride in LDS (breaks 40-dword stride)
#define THREADS    256         // 8 wave32 waves per block
#define F4_PER_ROW (CHUNK_N / 4)   // 128 float4 per block-row

// Clang-native 128-bit float vector (works with nontemporal builtins;
// HIP's float4 is a struct and is rejected by them).
typedef float v4f __attribute__((ext_vector_type(4)));

// Per-element spline evaluation. cp_row points at the padded LDS row for
// this neuron; the 4 gathered points are consecutive floats.
__device__ __forceinline__ float cr_eval(float xv, const float* __restrict__ cp_row) {
    // segment index: p0 = floor((x - (-2)) * (8-2)/4 + 1), clamped at range edges
    float s   = (xv + 2.0f) * 1.5f + 1.0f;
    float p0f = floorf(s);
    p0f = (xv <= -2.0f) ? 1.0f : p0f;
    p0f = (xv >=  2.0f) ? 7.0f : p0f;
    int p0 = (int)p0f;
    p0 = p0 < 1 ? 1 : (p0 > 7 ? 7 : p0);   // safety clamp, keeps gather in-row

    // local parameter u in [0,1):  t = x / delta_x,  delta_x = 0.5
    float t  = xv * 2.0f;
    float u  = t - floorf(t);
    float u2 = u * u;
    float u3 = u2 * u;

    // weights = reversed (U @ B) per reference:
    //   w(-1) = 0.5(u^3 - u^2)
    //   w( 0) = 0.5(-3u^3 + 4u^2 + u)
    //   w(+1) = 0.5( 3u^3 - 5u^2 + 2)
    //   w(+2) = 0.5(-u^3 + 2u^2 - u)
    float wm1 = 0.5f * (u3 - u2);
    float w0  = 0.5f * fmaf(-3.0f, u3, fmaf(4.0f, u2, u));
    float w1  = 0.5f * fmaf( 3.0f, u3, fmaf(-5.0f, u2, 2.0f));
    float w2  = 0.5f * fmaf(-1.0f, u3, fmaf(2.0f, u2, -u));

    float q0 = cp_row[p0 - 1];
    float q1 = cp_row[p0    ];
    float q2 = cp_row[p0 + 1];
    float q3 = cp_row[p0 + 2];

    return fmaf(wm1, q0, fmaf(w0, q1, fmaf(w1, q2, w2 * q3)));
}

__global__ __launch_bounds__(THREADS)
void catmullrom_act_kernel(const float* __restrict__ x,
                           const float* __restrict__ cp,
                           float* __restrict__ out) {
    __shared__ float s_cp[CHUNK_N * LDS_STRIDE];   // ~22 KB, padded rows

    const int c0  = blockIdx.x * CHUNK_N;   // first neuron of this block's chunk
    const int r0  = blockIdx.y * ROWS_PB;   // first row of this block's slab
    const int tid = threadIdx.x;

    // ---- Stage this chunk's control points into LDS (padded stride 11). ----
    // Global side is fully contiguous: cp[(c0+n)*10 + j] == cpsrc[i].
    const float* cpsrc = cp + (size_t)c0 * KCP;
    for (int i = tid; i < CHUNK_N * KCP; i += THREADS) {
        int n = i / KCP;            // constant divide -> mul/shift
        int j = i - n * KCP;
        s_cp[n * LDS_STRIDE + j] = cpsrc[i];
    }
    __syncthreads();

    // ---- Main streaming loop: 128-bit vectors along the neuron dimension. ----
    const int TOTAL = ROWS_PB * F4_PER_ROW;        // 4096 float4 per block
    for (int idx = tid; idx < TOTAL; idx += THREADS) {
        const int r = idx >> 7;                    // idx / F4_PER_ROW
        const int f = idx & (F4_PER_ROW - 1);      // float4 index within row

        const size_t base = (size_t)(r0 + r) * N_NEURONS + c0;
        const v4f* xp = reinterpret_cast<const v4f*>(x + base);
        v4f*       op = reinterpret_cast<v4f*>(out + base);

        // Non-temporal: x/out are stream-once; keep L2 free for the cp table.
        v4f xv = __builtin_nontemporal_load(xp + f);

        const int nl = f * 4;                      // local neuron of lane .x
        v4f o;
        o.x = cr_eval(xv.x, &s_cp[(nl + 0) * LDS_STRIDE]);
        o.y = cr_eval(xv.y, &s_cp[(nl + 1) * LDS_STRIDE]);
        o.z = cr_eval(xv.z, &s_cp[(nl + 2) * LDS_STRIDE]);
        o.w = cr_eval(xv.w, &s_cp[(nl + 3) * LDS_STRIDE]);

        __builtin_nontemporal_store(o, op + f);
    }
}

extern "C" void kernel_launch(void* const* d_in, const int* in_sizes, int n_in,
                              void* d_out, int out_size, void* d_ws, size_t ws_size,
                              hipStream_t stream) {
    (void)in_sizes; (void)n_in; (void)d_ws; (void)ws_size; (void)out_size;

    const float* x  = (const float*)d_in[0];   // (8192, 2048) f32
    const float* cp = (const float*)d_in[1];   // (2048, 10)   f32
    float*       out = (float*)d_out;          // (8192, 2048) f32

    dim3 grid(N_NEURONS / CHUNK_N, M_ROWS / ROWS_PB);   // (4, 256) = 1024 blocks
    catmullrom_act_kernel<<<grid, dim3(THREADS), 0, stream>>>(x, cp, out);
}